// AttentionMessageWeighting_43533788512905
// MI455X (gfx1250) — compile-verified
//
#include <hip/hip_runtime.h>
#include <hip/hip_bf16.h>

typedef float v2f __attribute__((ext_vector_type(2)));
typedef float v8f __attribute__((ext_vector_type(8)));

#define DIM 128
#define NHEAD 8
#define NEG_SLOPE 0.01f

// Monotonic order-preserving float<->uint encoding so float-max == uint-max.
// Key 0 is below every real float's key, so it is the max-identity.
__device__ __forceinline__ unsigned enc_f32(float f) {
    unsigned u = __float_as_uint(f);
    return (u & 0x80000000u) ? ~u : (u | 0x80000000u);
}
__device__ __forceinline__ float dec_f32(unsigned k) {
    unsigned u = (k & 0x80000000u) ? (k & 0x7FFFFFFFu) : ~k;
    return __uint_as_float(u);
}

// Build per-lane WMMA B-operand slices (64 iterations x 32 lanes x float2)
// for the block-diagonal 256x16 weight matrix, and zero the segment tables.
// Bfull[c][n] (n<8): c<128 -> head c/16 gets weight[n][c%16];
//                    c>=128 -> head (c-128)/16 gets weight[n][16+(c-128)%16].
__global__ void init_kernel(const float* __restrict__ weight,
                            float2* __restrict__ B2,
                            unsigned* __restrict__ seg_max,
                            float* __restrict__ seg_sum,
                            int nseg) {
    int tid = blockIdx.x * blockDim.x + threadIdx.x;
    if (tid < 64 * 32) {
        int k = tid >> 5;          // WMMA iteration
        int L = tid & 31;          // lane
        int n = L & 15;            // output column (head)
        int col = 4 * k + 2 * (L >> 4);
        float bx = 0.f, by = 0.f;
        if (n < NHEAD) {
            int c0 = col, c1 = col + 1;
            int half0 = c0 >> 7, cc0 = c0 & 127;
            if ((cc0 >> 4) == n) bx = weight[n * 32 + half0 * 16 + (cc0 & 15)];
            int half1 = c1 >> 7, cc1 = c1 & 127;
            if ((cc1 >> 4) == n) by = weight[n * 32 + half1 * 16 + (cc1 & 15)];
        }
        B2[tid] = make_float2(bx, by);
    }
    if (tid < nseg) {
        seg_max[tid] = 0u;     // encodes "below -inf": max identity
        seg_sum[tid] = 0.0f;
    }
}

// Pass 1: one wave per 16 edges.
//  - fused copy of message rows to d_out (float4, coalesced)
//  - scores via 64x V_WMMA_F32_16X16X4_F32 (K=256 virtual GEMM)
//  - LeakyReLU, store raw scores to alpha region, atomic segment-max.
__global__ void __launch_bounds__(256) pass1_kernel(
    const int* __restrict__ edge_index, const float* __restrict__ message,
    const float* __restrict__ x_e, const float2* __restrict__ B2,
    float* __restrict__ out_msg, float* __restrict__ out_alpha,
    unsigned* __restrict__ seg_max, int E)
{
    int tid  = blockIdx.x * blockDim.x + threadIdx.x;
    int wave = tid >> 5;
    int L    = tid & 31;
    long long e0 = (long long)wave * 16;
    if (e0 >= E) return;

    int n = L & 15;
    long long eL = e0 + n;
    if (eL >= E) eL = E - 1;                       // clamp (E%16==0 normally)
    int tgt = edge_index[(long long)E + eL];       // target = edge_index[1][e]

    const float* vmsg = message + eL * DIM;
    const float* vx   = x_e + (long long)tgt * DIM;

    // Fused message -> d_out copy: 16 rows x 128 f32, lane L moves one float4/row.
    const float4* m4 = (const float4*)message;
    float4*       o4 = (float4*)out_msg;
#pragma unroll
    for (int r = 0; r < 16; ++r) {
        long long e = e0 + r;
        if (e < E) o4[e * (DIM / 4) + L] = m4[e * (DIM / 4) + L];
    }

    // A-operand (16x4 f32): lanes 0-15 hold (M=lane, K=0/1), lanes 16-31 (M=lane-16, K=2/3).
    v8f acc = {};
    int halfsel = (L >> 4) * 2;
#pragma unroll 4
    for (int k = 0; k < 64; ++k) {
        int col = 4 * k + halfsel;
        const float* src = (col < DIM) ? (vmsg + col) : (vx + (col - DIM));
        v2f a = *(const v2f*)src;                  // 8B aligned (col even)
        v2f b = *(const v2f*)&B2[k * 32 + L];
        acc = __builtin_amdgcn_wmma_f32_16x16x4_f32(
            false, a, false, b, (short)0, acc, false, false);
    }

    // D layout: VGPR i -> row M = 8*(L>=16)+i, column N = L&15.
    int mbase = (L >> 4) << 3;
#pragma unroll
    for (int i = 0; i < 8; ++i) {
        int M = mbase + i;
        float s = acc[i];
        s = (s >= 0.f) ? s : NEG_SLOPE * s;        // LeakyReLU
        int tg = __shfl(tgt, M, 32);               // all lanes execute (no divergence)
        long long e = e0 + M;
        if (n < NHEAD && e < E) {
            out_alpha[e * NHEAD + n] = s;
            atomicMax(&seg_max[(long long)tg * NHEAD + n], enc_f32(s));
        }
    }
}

// Pass 2: exp(s - seg_max), store back, atomic segment-sum.
__global__ void pass2_kernel(const int* __restrict__ edge_index,
                             float* __restrict__ alpha,
                             const unsigned* __restrict__ seg_max,
                             float* __restrict__ seg_sum, int E) {
    long long i = (long long)blockIdx.x * blockDim.x + threadIdx.x;
    if (i >= (long long)E * NHEAD) return;
    long long e = i >> 3;
    int h = (int)(i & 7);
    int tg = edge_index[(long long)E + e];
    float m = dec_f32(seg_max[(long long)tg * NHEAD + h]);
    float ex = expf(alpha[i] - m);
    alpha[i] = ex;
    atomicAdd(&seg_sum[(long long)tg * NHEAD + h], ex);
}

// Pass 3: normalize.
__global__ void pass3_kernel(const int* __restrict__ edge_index,
                             float* __restrict__ alpha,
                             const float* __restrict__ seg_sum, int E) {
    long long i = (long long)blockIdx.x * blockDim.x + threadIdx.x;
    if (i >= (long long)E * NHEAD) return;
    long long e = i >> 3;
    int h = (int)(i & 7);
    int tg = edge_index[(long long)E + e];
    alpha[i] = alpha[i] / (seg_sum[(long long)tg * NHEAD + h] + 1e-16f);
}

extern "C" void kernel_launch(void* const* d_in, const int* in_sizes, int n_in,
                              void* d_out, int out_size, void* d_ws, size_t ws_size,
                              hipStream_t stream) {
    const int*   edge_index = (const int*)d_in[0];   // (2,E) int32
    const float* message    = (const float*)d_in[1]; // (E,128)
    const float* x_e        = (const float*)d_in[2]; // (N,128)
    const float* weight     = (const float*)d_in[3]; // (8,32)

    int E = in_sizes[1] / DIM;
    int N = in_sizes[2] / DIM;
    int nseg = N * NHEAD;

    float* out_msg   = (float*)d_out;                       // E*128
    float* out_alpha = out_msg + (size_t)E * DIM;           // E*8

    char*     ws      = (char*)d_ws;
    float2*   B2      = (float2*)ws;                        // 64*32 float2 = 16KB
    unsigned* seg_max = (unsigned*)(ws + 16384);
    float*    seg_sum = (float*)(ws + 16384 + (size_t)nseg * sizeof(unsigned));

    int initN = (nseg > 64 * 32) ? nseg : 64 * 32;
    init_kernel<<<(initN + 255) / 256, 256, 0, stream>>>(weight, B2, seg_max, seg_sum, nseg);

    int waves  = (E + 15) / 16;
    int blocks = (waves + 7) / 8;                           // 8 waves (256 thr) / block
    pass1_kernel<<<blocks, 256, 0, stream>>>(edge_index, message, x_e, B2,
                                             out_msg, out_alpha, seg_max, E);

    long long tot = (long long)E * NHEAD;
    int b2 = (int)((tot + 255) / 256);
    pass2_kernel<<<b2, 256, 0, stream>>>(edge_index, out_alpha, seg_max, seg_sum, E);
    pass3_kernel<<<b2, 256, 0, stream>>>(edge_index, out_alpha, seg_sum, E);
}